// ConditionalRandomField_89455578841185
// MI455X (gfx1250) — compile-verified
//
#include <hip/hip_runtime.h>
#include <hip/hip_bf16.h>

#define B_ 1024
#define L_ 512
#define T_ 50
#define TP 64          // padded tag dim
#define BT 16          // batch tile per wave
#define KC 13          // K chunks needed (i = 0..51; rows >= 52 are all-zero)

typedef __attribute__((ext_vector_type(2))) float v2f;
typedef __attribute__((ext_vector_type(8))) float v8f;

// Half-wave swap (lane l <-> lane l^16) in the VALU via v_permlanex16_b32
// (identity selectors). Falls back to __shfl_xor (ds_bpermute) if absent.
__device__ __forceinline__ float xswap16(float x) {
#if __has_builtin(__builtin_amdgcn_permlanex16)
    int v = __builtin_amdgcn_permlanex16(__float_as_int(x), __float_as_int(x),
                                         0x76543210, 0xfedcba98,
                                         /*fi=*/true, /*bound_ctrl=*/false);
    return __int_as_float(v);
#else
    return __shfl_xor(x, 16, 32);
#endif
}

// ---------------- gold path score (embarrassingly parallel gather) ----------
__global__ void crf_gold_kernel(const float* __restrict__ feats,
                                const float* __restrict__ trans,
                                const float* __restrict__ start,
                                const float* __restrict__ endsc,
                                const long long* __restrict__ tags,
                                const int* __restrict__ lengths,
                                float* __restrict__ gold) {
    int b = blockIdx.x * blockDim.x + threadIdx.x;
    if (b >= B_) return;
    int len = lengths[b];
    const long long* tb = tags + (size_t)b * L_;
    const float* fb = feats + (size_t)b * L_ * T_;
    int prev = (int)tb[0];
    float g = start[prev] + fb[prev];      // t=0 emit always in-mask (len >= 1)
    int last = prev;
    for (int t = 1; t < L_; ++t) {
        int tag = (int)tb[t];
        if (t < len) {
            g += trans[prev * T_ + tag] + fb[(size_t)t * T_ + tag];
            last = tag;
        }
        prev = tag;
    }
    g += endsc[last];
    gold[b] = g;
}

// ---------------- forward algorithm via f32 WMMA ---------------------------
// One wave owns a 16-batch tile. alpha kept in WMMA C/D layout:
// element (j, b): tile jt = j/16, vgpr r = j%8, lane = b + 16*((j%16) >= 8).
// Per step: s[j,b] = sum_i E^T[j,i] * exp(alpha[i,b]-m[b])   (f32 WMMA GEMM)
//           alpha'[j,b] = m[b] + log(s[j,b]) + emit[b,t,j]   (masked select)
__global__ __launch_bounds__(32)
void crf_forward_kernel(const float* __restrict__ feats,
                        const float* __restrict__ trans,
                        const float* __restrict__ start,
                        const float* __restrict__ endsc,
                        const int* __restrict__ lengths,
                        const float* __restrict__ gold,
                        float* __restrict__ out) {
    __shared__ float sE[TP * TP];   // E[i][j] = exp(trans[i][j]), zero padded
    __shared__ float sStart[TP];
    __shared__ float sEnd[TP];

    const int lane  = threadIdx.x;        // 0..31
    const int bl    = lane & 15;          // batch within tile
    const int hi    = lane >> 4;          // half-wave select
    const int bBase = blockIdx.x * BT;

    // Build E = exp(trans), zero padded so padded tags ride at -inf forever.
    for (int idx = lane; idx < TP * TP; idx += 32) {
        int i = idx >> 6, j = idx & 63;
        float v = 0.0f;
        if (i < T_ && j < T_) v = __expf(trans[i * T_ + j]);
        sE[idx] = v;
    }
    for (int j = lane; j < TP; j += 32) {
        sStart[j] = (j < T_) ? start[j] : -1e30f;
        sEnd[j]   = (j < T_) ? endsc[j] : 0.0f;
    }
    __syncthreads();

    // Preload constant A fragments (A = E^T). Fragment (mt, c):
    // D rows j = 16mt..16mt+15, K rows i = 4c..4c+3.
    // f32 16x16x4 A layout: V0 = {K=0 | K=2 across half-waves}, V1 = {K=1 | K=3}.
    v2f aF[4][KC];
    #pragma unroll
    for (int mt = 0; mt < 4; ++mt) {
        #pragma unroll
        for (int c = 0; c < KC; ++c) {
            int i0 = 4 * c + 2 * hi;
            int j  = 16 * mt + bl;
            aF[mt][c].x = sE[(i0)     * TP + j];
            aF[mt][c].y = sE[(i0 + 1) * TP + j];
        }
    }

    const int myb = bBase + bl;
    const int len = lengths[myb];
    const float* fb = feats + (size_t)myb * L_ * T_;

    // alpha at t=0: emit + start (padded tags ~ -2e30, -> -inf after step 1)
    float alpha[4][8];
    #pragma unroll
    for (int jt = 0; jt < 4; ++jt)
        #pragma unroll
        for (int r = 0; r < 8; ++r) {
            int j = 16 * jt + 8 * hi + r;
            float e = (j < T_) ? fb[j] : -1e30f;
            alpha[jt][r] = e + sStart[j];
        }

    for (int t = 1; t < L_; ++t) {
        // ---- emit tile for step t in D layout (float2 loads, 8B aligned);
        //      issued first, consumed last -> latency hidden by the GEMM chain.
        float em[3][8];
        const float* ft = fb + (size_t)t * T_;
        #pragma unroll
        for (int jt = 0; jt < 3; ++jt) {
            int j0 = 16 * jt + 8 * hi;      // max j = 47 < 50, always valid
            #pragma unroll
            for (int rp = 0; rp < 4; ++rp) {
                float2 v = *(const float2*)(ft + j0 + 2 * rp);
                em[jt][2 * rp]     = v.x;
                em[jt][2 * rp + 1] = v.y;
            }
        }
        // tile 3: j=48,49 real; load unconditionally (in-bounds for all lanes),
        // select instead of branching on hi.
        float2 v3 = *(const float2*)(ft + 48);
        float em3[2];
        em3[0] = (hi == 0) ? v3.x : -1e30f;
        em3[1] = (hi == 0) ? v3.y : -1e30f;

        // ---- per-batch row max: per-lane max + one half-wave swap
        float m = -3.0e38f;
        #pragma unroll
        for (int jt = 0; jt < 4; ++jt)
            #pragma unroll
            for (int r = 0; r < 8; ++r) m = fmaxf(m, alpha[jt][r]);
        m = fmaxf(m, xswap16(m));

        // ---- p = exp(alpha - m); only rows 0..51 feed the GEMM (28 exps)
        float p[4][8];
        #pragma unroll
        for (int jt = 0; jt < 3; ++jt)
            #pragma unroll
            for (int r = 0; r < 8; ++r)
                p[jt][r] = __expf(alpha[jt][r] - m);
        #pragma unroll
        for (int r = 0; r < 4; ++r)
            p[3][r] = __expf(alpha[3][r] - m);

        // ---- assemble B fragments fully in-register:
        // chunk c rows 4c..4c+3 all live in half-wave g=(c>>1)&1; the other
        // half fetches them with one VALU half-swap, then select.
        v2f bF[KC];
        #pragma unroll
        for (int c = 0; c < KC; ++c) {
            const int a = c >> 2, d = c & 3, g = (d >> 1) & 1;
            const int rl = 4 * d - 6 * g;        // local base reg (when hi==g)
            const int rr = 4 * d + 2 - 10 * g;   // partner's reg (when hi!=g)
            float lx = p[a][rl],     ly = p[a][rl + 1];
            float rx = xswap16(p[a][rr]);
            float ry = xswap16(p[a][rr + 1]);
            bool sel = (hi == g);
            bF[c].x = sel ? lx : rx;
            bF[c].y = sel ? ly : ry;
        }

        // ---- GEMM: per jt tile, two independent WMMA chains (7+6) + add,
        //      halving the dependent-WMMA latency on the serial path.
        const bool valid = (t < len);
        #pragma unroll
        for (int jt = 0; jt < 4; ++jt) {
            v8f accA = {0.f, 0.f, 0.f, 0.f, 0.f, 0.f, 0.f, 0.f};
            v8f accB = {0.f, 0.f, 0.f, 0.f, 0.f, 0.f, 0.f, 0.f};
            #pragma unroll
            for (int c = 0; c < 7; ++c)
                accA = __builtin_amdgcn_wmma_f32_16x16x4_f32(
                           false, aF[jt][c], false, bF[c],
                           (short)0, accA, false, false);
            #pragma unroll
            for (int c = 7; c < KC; ++c)
                accB = __builtin_amdgcn_wmma_f32_16x16x4_f32(
                           false, aF[jt][c], false, bF[c],
                           (short)0, accB, false, false);
            #pragma unroll
            for (int r = 0; r < 8; ++r) {
                float s = accA[r] + accB[r];
                float e = (jt < 3) ? em[jt][r]
                                   : ((r < 2) ? em3[r] : -1e30f);
                float na = m + __logf(s) + e;
                alpha[jt][r] = valid ? na : alpha[jt][r];
            }
        }
    }

    // ---- finalize: add end scores, logsumexp over tags
    #pragma unroll
    for (int jt = 0; jt < 4; ++jt)
        #pragma unroll
        for (int r = 0; r < 8; ++r) {
            int j = 16 * jt + 8 * hi + r;
            alpha[jt][r] += sEnd[j];
        }
    float m = -3.0e38f;
    #pragma unroll
    for (int jt = 0; jt < 4; ++jt)
        #pragma unroll
        for (int r = 0; r < 8; ++r) m = fmaxf(m, alpha[jt][r]);
    m = fmaxf(m, xswap16(m));
    float s = 0.0f;
    #pragma unroll
    for (int jt = 0; jt < 4; ++jt)
        #pragma unroll
        for (int r = 0; r < 8; ++r) s += __expf(alpha[jt][r] - m);
    s += xswap16(s);
    float norm = m + __logf(s);

    if (hi == 0) out[myb] = norm - gold[myb];
}

// ---------------------------------------------------------------------------
extern "C" void kernel_launch(void* const* d_in, const int* in_sizes, int n_in,
                              void* d_out, int out_size, void* d_ws, size_t ws_size,
                              hipStream_t stream) {
    const float*     feats   = (const float*)d_in[0];
    const float*     trans   = (const float*)d_in[1];
    const float*     startsc = (const float*)d_in[2];
    const float*     endsc   = (const float*)d_in[3];
    const long long* tags    = (const long long*)d_in[4];   // int64 per reference
    const int*       lengths = (const int*)d_in[5];         // int32 per reference
    float* out  = (float*)d_out;
    float* gold = (float*)d_ws;                             // B floats scratch

    crf_gold_kernel<<<(B_ + 255) / 256, 256, 0, stream>>>(
        feats, trans, startsc, endsc, tags, lengths, gold);
    crf_forward_kernel<<<B_ / BT, 32, 0, stream>>>(
        feats, trans, startsc, endsc, lengths, gold, out);
}